// TridistributeGeneraotr_52501680226536
// MI455X (gfx1250) — compile-verified
//
#include <hip/hip_runtime.h>

// ---------------------------------------------------------------------------
// Trilinear 3D-LUT scatter-accumulate (DIM=33) for MI455X / gfx1250.
// Atomic-bound (L2-resident f32 atomics); loads staged via CDNA5 async
// global->LDS copies (ASYNCcnt) with per-wave double buffering.
// ---------------------------------------------------------------------------

#define DIM    33
#define NBINS  (DIM * DIM * DIM)        // 35937
#define NB     8
#define PLANE  (512 * 512)              // 262144 = 1<<18
#define PSHIFT 18
#define NPIX   (NB * PLANE)             // 2097152
#define LUTSZ  (NB * 3 * NBINS)         // 862488 floats
#define OUTSZ  (NB * 3 * PLANE)         // 6291456 floats
#define BINSIZE (1.000001f / 32.0f)

#define TILE   32                        // pixels per wave-tile (wave32)
#define NSTRM  7                         // mask + 3 in + 3 out
#define BUF_F  (NSTRM * TILE)            // 224 floats per buffer
#define WPB    8                         // waves per 256-thread block

#if defined(__has_builtin)
#  if __has_builtin(__builtin_amdgcn_global_load_async_to_lds_b32) && \
      __has_builtin(__builtin_amdgcn_s_wait_asynccnt)
#    define USE_ASYNC 1
#  endif
#endif
#ifndef USE_ASYNC
#  define USE_ASYNC 0
#endif

__device__ __forceinline__ void gatomic_add(float* p, float v) {
  // Agent-scope relaxed fp32 add -> global_atomic_add_f32 (no return).
  __hip_atomic_fetch_add(p, v, __ATOMIC_RELAXED, __HIP_MEMORY_SCOPE_AGENT);
}

#if USE_ASYNC
typedef __attribute__((address_space(1))) int gas_i32;   // global int
typedef __attribute__((address_space(3))) int lds_i32;   // LDS int

__device__ __forceinline__ void async_ld_f32(const float* g, float* l) {
  __builtin_amdgcn_global_load_async_to_lds_b32(
      (gas_i32*)(unsigned long long)(const void*)g,
      (lds_i32*)(unsigned int)(unsigned long long)(const void*)l,
      /*offset=*/0, /*cpol=*/0);
}
#endif

// Per-pixel trilinear scatter: 8 corners x (1 cnt + 3 lut) atomics.
__device__ __forceinline__ void scatter_pixel(
    float m, float r, float g, float b, float o0, float o1, float o2,
    int img, float* __restrict__ lut, float* __restrict__ cnt) {
  if (!(m > 0.0f)) return;   // mask==0 contributes exactly zero everywhere

  float xr = r / BINSIZE; int ir = (int)xr; float fr = xr - (float)ir;
  float xg = g / BINSIZE; int ig = (int)xg; float fg = xg - (float)ig;
  float xb = b / BINSIZE; int ib = (int)xb; float fb = xb - (float)ib;
  int base = ir + ig * DIM + ib * DIM * DIM;

  float wr[2] = {1.0f - fr, fr};
  float wg[2] = {1.0f - fg, fg};
  float wb[2] = {1.0f - fb, fb};

  float* lb = lut + (size_t)img * 3 * NBINS;   // channel planes at +c*NBINS
  float* cb = cnt + (size_t)img * 3 * NBINS;   // cnt accumulated in c=0 slice

#pragma unroll
  for (int db = 0; db < 2; ++db)
#pragma unroll
    for (int dg = 0; dg < 2; ++dg)
#pragma unroll
      for (int dr = 0; dr < 2; ++dr) {
        float w = wr[dr] * wg[dg] * wb[db];
        int bin = base + dr + dg * DIM + db * (DIM * DIM);
        gatomic_add(cb + bin, w);
        gatomic_add(lb + bin, w * o0);
        gatomic_add(lb + NBINS + bin, w * o1);
        gatomic_add(lb + 2 * NBINS + bin, w * o2);
      }
}

__global__ __launch_bounds__(256) void tridist_scatter_kernel(
    const float* __restrict__ mask, const float* __restrict__ inp,
    const float* __restrict__ outp, float* __restrict__ lut,
    float* __restrict__ cnt) {
  const int lane   = threadIdx.x & 31;
  const int gwave  = (int)((blockIdx.x * blockDim.x + threadIdx.x) >> 5);
  const int nwave  = (int)((gridDim.x * blockDim.x) >> 5);
  const int ntiles = NPIX / TILE;      // 65536, PLANE divisible by TILE

#if USE_ASYNC
  __shared__ float smem[WPB * 2 * BUF_F];       // 14.3 KB
  float* wbuf = &smem[(threadIdx.x >> 5) * 2 * BUF_F];

  auto prefetch = [&](int tile, int k) {
    int p0 = tile * TILE;
    int img = p0 >> PSHIFT;
    int i   = (p0 & (PLANE - 1)) + lane;
    float* dst = wbuf + k * BUF_F + lane;
    const float* mb = mask + ((size_t)img << PSHIFT) + i;
    const float* ib = inp  + ((size_t)img * 3 << PSHIFT) + i;
    const float* ob = outp + ((size_t)img * 3 << PSHIFT) + i;
    async_ld_f32(mb,             dst + 0 * TILE);
    async_ld_f32(ib,             dst + 1 * TILE);
    async_ld_f32(ib + PLANE,     dst + 2 * TILE);
    async_ld_f32(ib + 2 * PLANE, dst + 3 * TILE);
    async_ld_f32(ob,             dst + 4 * TILE);
    async_ld_f32(ob + PLANE,     dst + 5 * TILE);
    async_ld_f32(ob + 2 * PLANE, dst + 6 * TILE);
  };

  int tile = gwave;
  int k = 0;
  if (tile < ntiles) prefetch(tile, 0);
  for (; tile < ntiles; tile += nwave) {
    int next = tile + nwave;
    if (next < ntiles) {
      prefetch(next, k ^ 1);
      __builtin_amdgcn_s_wait_asynccnt(NSTRM);   // current tile's 7 done
    } else {
      __builtin_amdgcn_s_wait_asynccnt(0);
    }
    const float* buf = wbuf + k * BUF_F + lane;
    int img = (tile * TILE) >> PSHIFT;
    scatter_pixel(buf[0 * TILE], buf[1 * TILE], buf[2 * TILE], buf[3 * TILE],
                  buf[4 * TILE], buf[5 * TILE], buf[6 * TILE], img, lut, cnt);
    k ^= 1;
  }
#else
  for (int tile = gwave; tile < ntiles; tile += nwave) {
    int p0 = tile * TILE;
    int img = p0 >> PSHIFT;
    int i   = (p0 & (PLANE - 1)) + lane;
    const float* mb = mask + ((size_t)img << PSHIFT) + i;
    const float* ib = inp  + ((size_t)img * 3 << PSHIFT) + i;
    const float* ob = outp + ((size_t)img * 3 << PSHIFT) + i;
    scatter_pixel(mb[0], ib[0], ib[PLANE], ib[2 * PLANE],
                  ob[0], ob[PLANE], ob[2 * PLANE], img, lut, cnt);
  }
#endif
}

__global__ void tridist_zero_kernel(float4* __restrict__ p, int n4) {
  int stride = (int)(gridDim.x * blockDim.x);
  for (int i = (int)(blockIdx.x * blockDim.x + threadIdx.x); i < n4; i += stride)
    p[i] = make_float4(0.f, 0.f, 0.f, 0.f);
}

__global__ void tridist_finalize_kernel(float* __restrict__ lut,
                                        float* __restrict__ cnt) {
  int t = (int)(blockIdx.x * blockDim.x + threadIdx.x);
  if (t >= NB * NBINS) return;
  int img = t / NBINS;
  int bin = t - img * NBINS;
  size_t base = (size_t)img * 3 * NBINS + bin;
  float c = cnt[base];
  bool nz = c > 0.0f;
#pragma unroll
  for (int ch = 0; ch < 3; ++ch) {
    size_t idx = base + (size_t)ch * NBINS;
    float l = lut[idx];
    lut[idx] = nz ? (l / c) : 0.0f;
    cnt[idx] = c;                      // broadcast cnt across channels
  }
}

__global__ void tridist_copy_kernel(const float4* __restrict__ src,
                                    float4* __restrict__ dst, int n4) {
  int stride = (int)(gridDim.x * blockDim.x);
  for (int i = (int)(blockIdx.x * blockDim.x + threadIdx.x); i < n4; i += stride)
    dst[i] = src[i];
}

extern "C" void kernel_launch(void* const* d_in, const int* in_sizes, int n_in,
                              void* d_out, int out_size, void* d_ws, size_t ws_size,
                              hipStream_t stream) {
  (void)in_sizes; (void)n_in; (void)d_ws; (void)ws_size; (void)out_size;
  const float* mask = (const float*)d_in[0];
  const float* inp  = (const float*)d_in[1];
  const float* outp = (const float*)d_in[2];

  float* out   = (float*)d_out;
  float* lut   = out;                 // [8,3,35937]
  float* cnt   = out + LUTSZ;         // [8,3,35937]
  float* ocopy = out + 2 * LUTSZ;     // [8,3,512,512]

  // 1) zero accumulators (lut + cnt regions), vectorized
  {
    int n4 = (2 * LUTSZ) / 4;         // 431244
    int blocks = (n4 + 255) / 256;
    tridist_zero_kernel<<<blocks, 256, 0, stream>>>((float4*)lut, n4);
  }
  // 2) scatter-accumulate (atomic-bound)
  tridist_scatter_kernel<<<2048, 256, 0, stream>>>(mask, inp, outp, lut, cnt);
  // 3) normalize + broadcast cnt
  {
    int n = NB * NBINS;               // 287496
    int blocks = (n + 255) / 256;
    tridist_finalize_kernel<<<blocks, 256, 0, stream>>>(lut, cnt);
  }
  // 4) pass-through copy of `output`
  {
    int n4 = OUTSZ / 4;               // 1572864
    tridist_copy_kernel<<<2048, 256, 0, stream>>>((const float4*)outp,
                                                  (float4*)ocopy, n4);
  }
}